// reliability_based_co_teaching_loss_J_26955214750391
// MI455X (gfx1250) — compile-verified
//
#include <hip/hip_runtime.h>
#include <math.h>

typedef __attribute__((ext_vector_type(2))) float v2f;
typedef __attribute__((ext_vector_type(8))) float v8f;

#define HW        262144      // 512*512 pixels per batch image
#define BB        4           // batch
#define CC        4           // classes
#define DD        32          // feature dim
#define EPSV      1e-8f
#define NWG2      256         // K2 workgroups per batch
#define WAVES     8           // 256 threads / wave32
#define K4_WGX    512         // K4 workgroups per batch
#define PART_STRIDE 264       // per (b,wg): 2 maps * (4*32 sums + 4 counts)

// ---------------------------------------------------------------------------
// K2: confidence-weighted center sums via V_WMMA_F32_16X16X4_F32.
//   sums[c,d] += onehot(hard==c)*conf * pseudo_feat[d]   (both maps)
// WMMA mapping: M=class (rows 4..15 zero), N=d (two accs: d 0..15 / 16..31),
// K=4 pixels per WMMA. Wave handles 128 contiguous pixels.
// ---------------------------------------------------------------------------
__global__ void k2_center_sums(const float* __restrict__ pm,
                               const float* __restrict__ pa,
                               const float* __restrict__ fm,
                               const float* __restrict__ fa,
                               float* __restrict__ part1) {
  const int b    = blockIdx.y;
  const int wg   = blockIdx.x;
  const int t    = threadIdx.x;
  const int wave = t >> 5;
  const int lane = t & 31;
  const int m    = lane & 15;   // matrix row (class, only 0..3 live)
  const int hi   = lane >> 4;   // K-half: pixels 2*hi, 2*hi+1

  const size_t boP = (size_t)b * CC * HW;
  const size_t boF = (size_t)b * DD * HW;

  const int waveGlobal = wg * WAVES + wave;
  const int pixPerWave = HW / (NWG2 * WAVES);   // 128
  const int pstart     = waveGlobal * pixPerWave;

  v8f accM0 = {}, accM1 = {}, accA0 = {}, accA1 = {};
  float cntM = 0.f, cntA = 0.f;

  for (int p0 = pstart; p0 < pstart + pixPerWave; p0 += 4) {
    // lanes 0..3 compute per-pixel head stats for pixels p0+lane
    int hm = 0, ha = 0;
    float cm = 0.f, ca = 0.f;
    if (lane < 4) {
      const int p = p0 + lane;
      float x0 = pm[boP + 0 * (size_t)HW + p];
      float x1 = pm[boP + 1 * (size_t)HW + p];
      float x2 = pm[boP + 2 * (size_t)HW + p];
      float x3 = pm[boP + 3 * (size_t)HW + p];
      float mx = x0; hm = 0;
      if (x1 > mx) { mx = x1; hm = 1; }
      if (x2 > mx) { mx = x2; hm = 2; }
      if (x3 > mx) { mx = x3; hm = 3; }
      float s = __expf(x0 - mx) + __expf(x1 - mx) + __expf(x2 - mx) + __expf(x3 - mx);
      cm = 1.f / s;   // max softmax prob

      float y0 = pa[boP + 0 * (size_t)HW + p];
      float y1 = pa[boP + 1 * (size_t)HW + p];
      float y2 = pa[boP + 2 * (size_t)HW + p];
      float y3 = pa[boP + 3 * (size_t)HW + p];
      float my = y0; ha = 0;
      if (y1 > my) { my = y1; ha = 1; }
      if (y2 > my) { my = y2; ha = 2; }
      if (y3 > my) { my = y3; ha = 3; }
      float sy = __expf(y0 - my) + __expf(y1 - my) + __expf(y2 - my) + __expf(y3 - my);
      ca = 1.f / sy;
    }
    // broadcast stats of this lane's two K pixels
    const int k0 = hi * 2, k1 = k0 + 1;
    int   hm0 = __shfl(hm, k0, 32), hm1 = __shfl(hm, k1, 32);
    float cm0 = __shfl(cm, k0, 32), cm1 = __shfl(cm, k1, 32);
    int   ha0 = __shfl(ha, k0, 32), ha1 = __shfl(ha, k1, 32);
    float ca0 = __shfl(ca, k0, 32), ca1 = __shfl(ca, k1, 32);

    const float* s0 = (cm0 >= ca0) ? fm : fa;   // pseudo-feature select
    const float* s1 = (cm1 >= ca1) ? fm : fa;
    const int pA = p0 + k0, pB = p0 + k1;

    v2f am, aa, b0, b1;
    am.x = (hm0 == m) ? cm0 : 0.f;  am.y = (hm1 == m) ? cm1 : 0.f;
    aa.x = (ha0 == m) ? ca0 : 0.f;  aa.y = (ha1 == m) ? ca1 : 0.f;
    b0.x = s0[boF + (size_t)m * HW + pA];
    b0.y = s1[boF + (size_t)m * HW + pB];
    b1.x = s0[boF + (size_t)(m + 16) * HW + pA];
    b1.y = s1[boF + (size_t)(m + 16) * HW + pB];

    accM0 = __builtin_amdgcn_wmma_f32_16x16x4_f32(false, am, false, b0, (short)0, accM0, false, false);
    accM1 = __builtin_amdgcn_wmma_f32_16x16x4_f32(false, am, false, b1, (short)0, accM1, false, false);
    accA0 = __builtin_amdgcn_wmma_f32_16x16x4_f32(false, aa, false, b0, (short)0, accA0, false, false);
    accA1 = __builtin_amdgcn_wmma_f32_16x16x4_f32(false, aa, false, b1, (short)0, accA1, false, false);

    cntM += (float)((hm0 == m) + (hm1 == m));
    cntA += (float)((ha0 == m) + (ha1 == m));
  }

  // deterministic per-workgroup reduction: per-wave slots, then fixed order
  __shared__ float lds_part[WAVES][2][CC][DD];   // 8 KB
  __shared__ float lds_cnt[WAVES][2][32];        // 2 KB
  if (lane < 16) {
#pragma unroll
    for (int r = 0; r < 4; ++r) {
      // D layout: VGPR r, lanes 0..15 hold M=r, N=lane (acc1 covers d+16)
      lds_part[wave][0][r][lane]      = accM0[r];
      lds_part[wave][0][r][lane + 16] = accM1[r];
      lds_part[wave][1][r][lane]      = accA0[r];
      lds_part[wave][1][r][lane + 16] = accA1[r];
    }
  }
  lds_cnt[wave][0][lane] = cntM;
  lds_cnt[wave][1][lane] = cntA;
  __syncthreads();

  float* dst = part1 + (size_t)(b * NWG2 + wg) * PART_STRIDE;
  {
    const int map = t >> 7, c = (t >> 5) & 3, d = t & 31;
    float s = 0.f;
#pragma unroll
    for (int w = 0; w < WAVES; ++w) s += lds_part[w][map][c][d];
    dst[map * 132 + c * 32 + d] = s;
  }
  if (t < 8) {
    const int map = t >> 2, c = t & 3;
    float s = 0.f;
#pragma unroll
    for (int w = 0; w < WAVES; ++w) s += lds_cnt[w][map][c] + lds_cnt[w][map][c + 16];
    dst[map * 132 + 128 + c] = s;
  }
}

// ---------------------------------------------------------------------------
// K3: reduce per-wg partials -> centers (/max(count,1)) + center norms.
// grid.x = batch, 256 threads: t -> (map, class, d). Fixed-order sums.
// ---------------------------------------------------------------------------
__global__ void k3_centers(const float* __restrict__ part1,
                           float* __restrict__ centers,
                           float* __restrict__ cnorm) {
  const int b = blockIdx.x;
  const int t = threadIdx.x;
  const int map = t >> 7, c = (t >> 5) & 3, d = t & 31;
  float s = 0.f, cnt = 0.f;
  for (int wg = 0; wg < NWG2; ++wg) {
    const float* base = part1 + (size_t)(b * NWG2 + wg) * PART_STRIDE + map * 132;
    s   += base[c * 32 + d];
    cnt += base[128 + c];
  }
  const float center = s / fmaxf(cnt, 1.0f);
  centers[(((size_t)b * 2 + map) * CC + c) * DD + d] = center;

  __shared__ float sq[256];
  sq[t] = center * center;
  __syncthreads();
  if (d == 0) {
    float n2 = 0.f;
#pragma unroll
    for (int j = 0; j < 32; ++j) n2 += sq[t + j];
    cnorm[((size_t)b * 2 + map) * CC + c] = sqrtf(n2);
  }
}

// ---------------------------------------------------------------------------
// K4: per-pixel reliability + CE, global weighted sums (block partials).
// lane<->pixel for coalesced feature reads; centers cached in LDS.
// ---------------------------------------------------------------------------
__global__ void k4_loss_terms(const float* __restrict__ pm,
                              const float* __restrict__ pa,
                              const float* __restrict__ fm,
                              const float* __restrict__ fa,
                              const float* __restrict__ centers,
                              const float* __restrict__ cnorm,
                              float* __restrict__ part4) {
  const int b = blockIdx.y;
  const int t = threadIdx.x;

  __shared__ float cM[CC][DD], cA[CC][DD];
  __shared__ float nM[CC], nA[CC];
  {
    const int map = t >> 7, c = (t >> 5) & 3, d = t & 31;
    const float v = centers[(((size_t)b * 2 + map) * CC + c) * DD + d];
    if (map == 0) cM[c][d] = v; else cA[c][d] = v;
    if (t < 8) {
      const int mp = t >> 2, ccl = t & 3;
      const float nv = cnorm[((size_t)b * 2 + mp) * CC + ccl];
      if (mp == 0) nM[ccl] = nv; else nA[ccl] = nv;
    }
  }
  __syncthreads();

  const size_t boP = (size_t)b * CC * HW;
  const size_t boF = (size_t)b * DD * HW;

  float a0 = 0.f, a1 = 0.f, a2 = 0.f, a3 = 0.f;
  for (int p = blockIdx.x * blockDim.x + t; p < HW; p += K4_WGX * 256) {
    float x0 = pm[boP + 0 * (size_t)HW + p];
    float x1 = pm[boP + 1 * (size_t)HW + p];
    float x2 = pm[boP + 2 * (size_t)HW + p];
    float x3 = pm[boP + 3 * (size_t)HW + p];
    float mxM = x0; int hm = 0;
    if (x1 > mxM) { mxM = x1; hm = 1; }
    if (x2 > mxM) { mxM = x2; hm = 2; }
    if (x3 > mxM) { mxM = x3; hm = 3; }
    float sM = __expf(x0 - mxM) + __expf(x1 - mxM) + __expf(x2 - mxM) + __expf(x3 - mxM);
    float cm = 1.f / sM;

    float y0 = pa[boP + 0 * (size_t)HW + p];
    float y1 = pa[boP + 1 * (size_t)HW + p];
    float y2 = pa[boP + 2 * (size_t)HW + p];
    float y3 = pa[boP + 3 * (size_t)HW + p];
    float mxA = y0; int ha = 0;
    if (y1 > mxA) { mxA = y1; ha = 1; }
    if (y2 > mxA) { mxA = y2; ha = 2; }
    if (y3 > mxA) { mxA = y3; ha = 3; }
    float sA = __expf(y0 - mxA) + __expf(y1 - mxA) + __expf(y2 - mxA) + __expf(y3 - mxA);
    float ca = 1.f / sA;

    const bool mask = (cm >= ca);
    const int pl = mask ? hm : ha;
    const float xp = (pl == 0) ? x0 : (pl == 1) ? x1 : (pl == 2) ? x2 : x3;
    const float yp = (pl == 0) ? y0 : (pl == 1) ? y1 : (pl == 2) ? y2 : y3;
    const float ceM = mxM + __logf(sM) - xp;   // -log_softmax picked
    const float ceA = mxA + __logf(sA) - yp;

    const float* fsel = mask ? fm : fa;        // read only the selected head
    float fn2 = 0.f, dM = 0.f, dA = 0.f;
#pragma unroll
    for (int d = 0; d < DD; ++d) {
      const float v = fsel[boF + (size_t)d * HW + p];
      fn2 += v * v;
      dM  += v * cM[hm][d];
      dA  += v * cA[ha][d];
    }
    const float fn   = fmaxf(sqrtf(fn2), EPSV);
    const float relM = dM / (fn * fmaxf(nM[hm], EPSV));
    const float relA = dA / (fn * fmaxf(nA[ha], EPSV));

    a0 += ceM * relA;   // loss_main numerator
    a1 += relA;         // loss_main denominator
    a2 += ceA * relM;   // loss_aux numerator
    a3 += relM;         // loss_aux denominator
  }

  // deterministic in-block tree reductions
  __shared__ float red[256];
  float acc[4] = {a0, a1, a2, a3};
#pragma unroll
  for (int q = 0; q < 4; ++q) {
    __syncthreads();
    red[t] = acc[q];
    __syncthreads();
    for (int s = 128; s > 0; s >>= 1) {
      if (t < s) red[t] += red[t + s];
      __syncthreads();
    }
    if (t == 0) part4[(size_t)(b * K4_WGX + blockIdx.x) * 4 + q] = red[0];
  }
}

// ---------------------------------------------------------------------------
// K5: fixed-order final reduce -> scalar loss.
// ---------------------------------------------------------------------------
__global__ void k5_finalize(const float* __restrict__ part4, float* __restrict__ out) {
  __shared__ float s[4];
  const int t = threadIdx.x;
  if (t < 4) {
    float acc = 0.f;
    for (int j = 0; j < BB * K4_WGX; ++j) acc += part4[(size_t)j * 4 + t];
    s[t] = acc;
  }
  __syncthreads();
  if (t == 0) out[0] = s[0] / s[1] + s[2] / s[3];
}

// ---------------------------------------------------------------------------
extern "C" void kernel_launch(void* const* d_in, const int* in_sizes, int n_in,
                              void* d_out, int out_size, void* d_ws, size_t ws_size,
                              hipStream_t stream) {
  const float* pm = (const float*)d_in[0];
  const float* pa = (const float*)d_in[1];
  const float* fm = (const float*)d_in[2];
  const float* fa = (const float*)d_in[3];

  float* ws      = (float*)d_ws;
  float* part1   = ws;                                        // 4*256*264 floats
  float* centers = part1 + (size_t)BB * NWG2 * PART_STRIDE;   // 4*2*4*32
  float* cnorm   = centers + (size_t)BB * 2 * CC * DD;        // 4*2*4
  float* part4   = cnorm + (size_t)BB * 2 * CC;               // 4*512*4
  float* out     = (float*)d_out;

  dim3 blk(256);
  k2_center_sums<<<dim3(NWG2, BB), blk, 0, stream>>>(pm, pa, fm, fa, part1);
  k3_centers    <<<dim3(BB),       blk, 0, stream>>>(part1, centers, cnorm);
  k4_loss_terms <<<dim3(K4_WGX, BB), blk, 0, stream>>>(pm, pa, fm, fa, centers, cnorm, part4);
  k5_finalize   <<<dim3(1), dim3(64), 0, stream>>>(part4, out);
}